// RNA2vec_GCN_2layer_76922864271371
// MI455X (gfx1250) — compile-verified
//
#include <hip/hip_runtime.h>
#include <math.h>

typedef __attribute__((ext_vector_type(2))) float v2f;
typedef __attribute__((ext_vector_type(8))) float v8f;

#define B_   128
#define N_   1024
#define EP_  8192
#define E_   (B_ * EP_)
#define K1_  820
#define K2_  656
#define H_   120
#define T_   19

__device__ __forceinline__ float sigm_(float x) { return 1.0f / (1.0f + expf(-x)); }

__device__ __forceinline__ v8f wmma_k4(v2f a, v2f b, v8f c) {
  return __builtin_amdgcn_wmma_f32_16x16x4_f32(false, a, false, b, (short)0, c, false, false);
}

// ---------------- GraphConv layer 1: out = relu(agg@Wrel + x@Wroot + b), K=4 ----------------
// One wave computes a full 16x128 row-block: A frags loaded once, 16 WMMAs.
__global__ void k_gconv1(const float* __restrict__ agg, const float* __restrict__ x,
                         const float* __restrict__ Wrel, const float* __restrict__ Wroot,
                         const float* __restrict__ bias, float* __restrict__ out, int M) {
  int wave = blockIdx.x * (blockDim.x >> 5) + (threadIdx.x >> 5);
  int mt = M >> 4;
  if (wave >= mt) return;
  int tm = wave;
  int lane = threadIdx.x & 31;
  int half = lane >> 4, lr = lane & 15;
  int m  = tm * 16 + lr;           // A-row held by this lane
  int ka = half * 2;               // K offset (0 or 2)

  v2f a0; a0.x = agg[(size_t)m * 4 + ka]; a0.y = agg[(size_t)m * 4 + ka + 1];
  v2f a1; a1.x = x[(size_t)m * 4 + ka];   a1.y = x[(size_t)m * 4 + ka + 1];

  v8f acc[8];
#pragma unroll
  for (int n = 0; n < 8; ++n) acc[n] = (v8f){};

#pragma unroll
  for (int n = 0; n < 8; ++n) {
    int col = n * 16 + lr;
    v2f b0; b0.x = Wrel[(size_t)ka * 128 + col];  b0.y = Wrel[(size_t)(ka + 1) * 128 + col];
    v2f b1; b1.x = Wroot[(size_t)ka * 128 + col]; b1.y = Wroot[(size_t)(ka + 1) * 128 + col];
    acc[n] = wmma_k4(a0, b0, acc[n]);
    acc[n] = wmma_k4(a1, b1, acc[n]);
  }

#pragma unroll
  for (int n = 0; n < 8; ++n) {
    int col = n * 16 + lr;
    float bb = bias[col];
#pragma unroll
    for (int r = 0; r < 8; ++r) {
      int mm = tm * 16 + r + half * 8;
      float v = acc[n][r] + bb;
      out[(size_t)mm * 128 + col] = v > 0.f ? v : 0.f;
    }
  }
}

// ---------------- GraphConv layer 2: K=128; one wave -> 16x128 row-block ----------------
__global__ void k_gconv2(const float* __restrict__ agg, const float* __restrict__ xr,
                         const float* __restrict__ Wrel, const float* __restrict__ Wroot,
                         const float* __restrict__ bias, float* __restrict__ out, int M) {
  int wave = blockIdx.x * (blockDim.x >> 5) + (threadIdx.x >> 5);
  int mt = M >> 4;
  if (wave >= mt) return;
  int tm = wave;
  int lane = threadIdx.x & 31;
  int half = lane >> 4, lr = lane & 15;
  int m = tm * 16 + lr;

  v8f acc[8];
#pragma unroll
  for (int n = 0; n < 8; ++n) acc[n] = (v8f){};

  const float* arow = agg + (size_t)m * 128;
  for (int k0 = 0; k0 < 128; k0 += 4) {
    int ka = k0 + half * 2;
    v2f a; a.x = arow[ka]; a.y = arow[ka + 1];
#pragma unroll
    for (int n = 0; n < 8; ++n) {
      int col = n * 16 + lr;
      v2f b; b.x = Wrel[(size_t)ka * 128 + col]; b.y = Wrel[(size_t)(ka + 1) * 128 + col];
      acc[n] = wmma_k4(a, b, acc[n]);
    }
  }
  const float* xrow = xr + (size_t)m * 128;
  for (int k0 = 0; k0 < 128; k0 += 4) {
    int ka = k0 + half * 2;
    v2f a; a.x = xrow[ka]; a.y = xrow[ka + 1];
#pragma unroll
    for (int n = 0; n < 8; ++n) {
      int col = n * 16 + lr;
      v2f b; b.x = Wroot[(size_t)ka * 128 + col]; b.y = Wroot[(size_t)(ka + 1) * 128 + col];
      acc[n] = wmma_k4(a, b, acc[n]);
    }
  }

#pragma unroll
  for (int n = 0; n < 8; ++n) {
    int col = n * 16 + lr;
    float bb = bias[col];
#pragma unroll
    for (int r = 0; r < 8; ++r) {
      int mm = tm * 16 + r + half * 8;
      float v = acc[n][r] + bb;
      out[(size_t)mm * 128 + col] = v > 0.f ? v : 0.f;
    }
  }
}

// ---------------- edge scatter-adds ----------------
__global__ void k_scatter4(const float* __restrict__ x, const int* __restrict__ src,
                           const int* __restrict__ dst, float* __restrict__ agg) {
  size_t total = (size_t)E_ * 4;
  for (size_t i = (size_t)blockIdx.x * blockDim.x + threadIdx.x; i < total;
       i += (size_t)gridDim.x * blockDim.x) {
    size_t e = i >> 2; int f = (int)(i & 3);
    atomicAdd(&agg[(size_t)dst[e] * 4 + f], x[(size_t)src[e] * 4 + f]);
  }
}

__global__ void k_scatter128(const float* __restrict__ x, const int* __restrict__ src,
                             const int* __restrict__ dst, const int* __restrict__ valid,
                             float* __restrict__ agg) {
  size_t total = (size_t)E_ * 128;
  for (size_t i = (size_t)blockIdx.x * blockDim.x + threadIdx.x; i < total;
       i += (size_t)gridDim.x * blockDim.x) {
    size_t e = i >> 7; int f = (int)(i & 127);
    if (valid[e])
      atomicAdd(&agg[(size_t)dst[e] * 128 + f], x[(size_t)src[e] * 128 + f]);
  }
}

// ---------------- ||p|| ----------------
__global__ void k_pnorm(const float* __restrict__ p, float* __restrict__ out) {
  __shared__ float red[128];
  float v = p[threadIdx.x];
  red[threadIdx.x] = v * v;
  __syncthreads();
  for (int s = 64; s > 0; s >>= 1) {
    if ((int)threadIdx.x < s) red[threadIdx.x] += red[threadIdx.x + s];
    __syncthreads();
  }
  if (threadIdx.x == 0) out[0] = sqrtf(red[0]);
}

// ---------------- score = tanh(x.p/||p||) ----------------
__global__ void k_score(const float* __restrict__ x, const float* __restrict__ p,
                        const float* __restrict__ pn, float* __restrict__ score, int M) {
  int i = blockIdx.x * blockDim.x + threadIdx.x;
  if (i >= M) return;
  const float* xr = x + (size_t)i * 128;
  float s = 0.f;
#pragma unroll 8
  for (int j = 0; j < 128; ++j) s += xr[j] * p[j];
  score[i] = tanhf(s / pn[0]);
}

// ---------------- per-graph top-k via bitonic sort of 1024 (score,idx) ----------------
__global__ void k_topk(const float* __restrict__ score, int* __restrict__ perm,
                       int* __restrict__ mapping, int n_per, int k) {
  __shared__ float sc[1024];
  __shared__ int   id[1024];
  int g = blockIdx.x;
  for (int j = threadIdx.x; j < 1024; j += blockDim.x) {
    if (j < n_per) { sc[j] = score[(size_t)g * n_per + j]; id[j] = j; }
    else           { sc[j] = -__builtin_inff();           id[j] = j; }
  }
  __syncthreads();
  for (int ksz = 2; ksz <= 1024; ksz <<= 1) {
    for (int jj = ksz >> 1; jj > 0; jj >>= 1) {
      for (int idx = threadIdx.x; idx < 1024; idx += blockDim.x) {
        int ixj = idx ^ jj;
        if (ixj > idx) {
          bool desc = ((idx & ksz) == 0);
          float s0 = sc[idx], s1 = sc[ixj];
          int   i0 = id[idx], i1 = id[ixj];
          bool g0 = (s0 > s1) || (s0 == s1 && i0 < i1);   // e[idx] ranks higher
          bool g1 = (s1 > s0) || (s1 == s0 && i1 < i0);   // e[ixj] ranks higher
          if (desc ? g1 : g0) {
            sc[idx] = s1; sc[ixj] = s0;
            id[idx] = i1; id[ixj] = i0;
          }
        }
      }
      __syncthreads();
    }
  }
  for (int j = threadIdx.x; j < k; j += blockDim.x) {
    int node = id[j];
    int gold = g * n_per + node;
    int gnew = g * k + j;
    perm[gnew] = gold;
    if (mapping) mapping[gold] = gnew;
  }
}

// ---------------- new_x = x[perm] * score[perm] ----------------
__global__ void k_gather_scale(const float* __restrict__ x, const float* __restrict__ score,
                               const int* __restrict__ perm, float* __restrict__ out, int Mk) {
  int i = blockIdx.x * blockDim.x + threadIdx.x;
  if (i >= Mk * 128) return;
  int m = i >> 7, f = i & 127;
  int p = perm[m];
  out[i] = x[(size_t)p * 128 + f] * score[p];
}

// ---------------- readout: [max || mean] per graph -> xc[:,0:256] ----------------
__global__ void k_readout(const float* __restrict__ xp, float* __restrict__ xc,
                          int k, int accumulate) {
  int g = blockIdx.x, f = threadIdx.x;   // 128 threads
  const float* base = xp + (size_t)g * k * 128 + f;
  float mx = -3.402823466e38f, sm = 0.f;
  for (int i = 0; i < k; ++i) {
    float v = base[(size_t)i * 128];
    mx = fmaxf(mx, v);
    sm += v;
  }
  float mean = sm / (float)k;
  float* o = xc + (size_t)g * 4816;
  if (accumulate) { o[f] += mx; o[128 + f] += mean; }
  else            { o[f]  = mx; o[128 + f]  = mean; }
}

// ---------------- edge remap after pool1 ----------------
__global__ void k_remap(const int* __restrict__ src, const int* __restrict__ dst,
                        const int* __restrict__ mapping, int* __restrict__ s1,
                        int* __restrict__ d1, int* __restrict__ v1) {
  int e = blockIdx.x * blockDim.x + threadIdx.x;
  if (e >= E_) return;
  int ns = mapping[src[e]], nd = mapping[dst[e]];
  v1[e] = (ns >= 0) && (nd >= 0);
  s1[e] = ns < 0 ? 0 : ns;
  d1[e] = nd < 0 ? 0 : nd;
}

// ---------------- conv1d(k=3) + relu + avgpool(5) -> xs[T,B,128] ----------------
__global__ void k_convpool(const float* __restrict__ target, const float* __restrict__ Wc,
                           const float* __restrict__ bc, float* __restrict__ xs) {
  int i = blockIdx.x * blockDim.x + threadIdx.x;
  if (i >= T_ * 128 * 128) return;
  int o = i & 127; int tb = i >> 7; int b = tb & 127; int t = tb >> 7;
  const float* tg = target + (size_t)b * 30 * 101;
  const float* w  = Wc + (size_t)o * 90;
  float bias = bc[o];
  float acc = 0.f;
  for (int p = t * 5; p < t * 5 + 5; ++p) {
    float c = bias;
    for (int ci = 0; ci < 30; ++ci) {
      const float* tr = tg + ci * 101 + p;
      const float* wr = w + ci * 3;
      c += tr[0] * wr[0] + tr[1] * wr[1] + tr[2] * wr[2];
    }
    acc += fmaxf(c, 0.f);
  }
  xs[i] = acc * 0.2f;
}

// ---------------- gi = xs @ Wih^T + bih for all (t,b) ----------------
__global__ void k_gi(const float* __restrict__ xs, const float* __restrict__ Wih,
                     const float* __restrict__ bih, float* __restrict__ gi) {
  int i = blockIdx.x * blockDim.x + threadIdx.x;
  if (i >= T_ * 128 * 360) return;
  int j = i % 360; int tb = i / 360;
  const float* xr = xs + (size_t)tb * 128;
  const float* w  = Wih + (size_t)j * 128;
  float s = bih[j];
#pragma unroll 8
  for (int kk = 0; kk < 128; ++kk) s += xr[kk] * w[kk];
  gi[i] = s;
}

// ---------------- one GRU step, both directions ----------------
__global__ void k_gru_step(const float* __restrict__ gi_f, const float* __restrict__ gi_b,
                           const float* __restrict__ Whh_f, const float* __restrict__ Whh_b,
                           const float* __restrict__ bhh_f, const float* __restrict__ bhh_b,
                           const float* __restrict__ hin_f, const float* __restrict__ hin_b,
                           float* __restrict__ hout_f, float* __restrict__ hout_b,
                           float* __restrict__ xc, int t) {
  int i = blockIdx.x * blockDim.x + threadIdx.x;
  if (i >= 128 * H_) return;
  int dir = blockIdx.y;
  int b = i / H_, j = i % H_;
  int tt = dir ? (T_ - 1 - t) : t;
  const float* gi  = (dir ? gi_b : gi_f) + ((size_t)tt * 128 + b) * 360;
  const float* Whh = dir ? Whh_b : Whh_f;
  const float* bhh = dir ? bhh_b : bhh_f;
  const float* hin = (dir ? hin_b : hin_f) + (size_t)b * H_;
  float* hout      = (dir ? hout_b : hout_f) + (size_t)b * H_;
  float gr = bhh[j], gz = bhh[H_ + j], gn = bhh[2 * H_ + j];
  for (int kk = 0; kk < H_; ++kk) {
    float hv = hin[kk];
    gr += hv * Whh[(size_t)j * H_ + kk];
    gz += hv * Whh[(size_t)(H_ + j) * H_ + kk];
    gn += hv * Whh[(size_t)(2 * H_ + j) * H_ + kk];
  }
  float r  = sigm_(gi[j] + gr);
  float z  = sigm_(gi[H_ + j] + gz);
  float ng = tanhf(gi[2 * H_ + j] + r * gn);
  float hn = (1.f - z) * ng + z * hin[j];
  hout[j] = hn;
  xc[(size_t)b * 4816 + 256 + tt * 240 + dir * H_ + j] = hn;
}

// ---------------- dense1: relu(xc @ Wd1 + bd1) ----------------
__global__ void k_dense1(const float* __restrict__ xc, const float* __restrict__ Wd1,
                         const float* __restrict__ bd1, float* __restrict__ hid) {
  int i = blockIdx.x * blockDim.x + threadIdx.x;
  if (i >= 128 * 102) return;
  int b = i / 102, j = i % 102;
  const float* xr = xc + (size_t)b * 4816;
  float s = bd1[j];
  for (int kk = 0; kk < 4816; ++kk) s += xr[kk] * Wd1[(size_t)kk * 102 + j];
  hid[i] = s > 0.f ? s : 0.f;
}

// ---------------- dense3 + log_softmax ----------------
__global__ void k_out(const float* __restrict__ hid, const float* __restrict__ Wd3,
                      const float* __restrict__ bd3, float* __restrict__ out) {
  int b = blockIdx.x * blockDim.x + threadIdx.x;
  if (b >= 128) return;
  const float* h = hid + (size_t)b * 102;
  float l0 = bd3[0], l1 = bd3[1];
  for (int j = 0; j < 102; ++j) { l0 += h[j] * Wd3[j * 2]; l1 += h[j] * Wd3[j * 2 + 1]; }
  float m = fmaxf(l0, l1);
  float lse = m + logf(expf(l0 - m) + expf(l1 - m));
  out[b * 2]     = l0 - lse;
  out[b * 2 + 1] = l1 - lse;
}

extern "C" void kernel_launch(void* const* d_in, const int* in_sizes, int n_in,
                              void* d_out, int out_size, void* d_ws, size_t ws_size,
                              hipStream_t stream) {
  (void)in_sizes; (void)n_in; (void)out_size; (void)ws_size;
  const float* x       = (const float*)d_in[0];
  const float* target  = (const float*)d_in[1];
  const int*   esrc    = (const int*)d_in[2];
  const int*   edst    = (const int*)d_in[3];
  const float* W1_rel  = (const float*)d_in[4];
  const float* W1_root = (const float*)d_in[5];
  const float* b1      = (const float*)d_in[6];
  const float* p1      = (const float*)d_in[7];
  const float* W2_rel  = (const float*)d_in[8];
  const float* W2_root = (const float*)d_in[9];
  const float* b2      = (const float*)d_in[10];
  const float* p2      = (const float*)d_in[11];
  const float* Wc      = (const float*)d_in[12];
  const float* bc      = (const float*)d_in[13];
  const float* Wih_f   = (const float*)d_in[14];
  const float* Whh_f   = (const float*)d_in[15];
  const float* bih_f   = (const float*)d_in[16];
  const float* bhh_f   = (const float*)d_in[17];
  const float* Wih_b   = (const float*)d_in[18];
  const float* Whh_b   = (const float*)d_in[19];
  const float* bih_b   = (const float*)d_in[20];
  const float* bhh_b   = (const float*)d_in[21];
  const float* Wd1     = (const float*)d_in[22];
  const float* bd1     = (const float*)d_in[23];
  const float* Wd3     = (const float*)d_in[24];
  const float* bd3     = (const float*)d_in[25];
  float* out = (float*)d_out;

  const int M1 = B_ * N_;    // 131072
  const int M2 = B_ * K1_;   // 104960
  const int M3 = B_ * K2_;   // 83968

  char* ws = (char*)d_ws;
  size_t off = 0;
  auto alloc = [&](size_t bytes) -> char* {
    char* p = ws + off;
    off = (off + bytes + 255) & ~(size_t)255;
    return p;
  };

  // region A: h1 / agg2 / h2p (lifetimes disjoint)
  float* regionA = (float*)alloc((size_t)M1 * 128 * 4);
  float* h1 = regionA, *agg2 = regionA, *h2p = regionA;
  float* h2       = (float*)alloc((size_t)M2 * 128 * 4);
  float* h1p      = (float*)alloc((size_t)M2 * 128 * 4);
  float* agg1     = (float*)alloc((size_t)M1 * 4 * 4);
  float* score1   = (float*)alloc((size_t)M1 * 4);
  float* score2   = (float*)alloc((size_t)M2 * 4);
  int*   perm1    = (int*)alloc((size_t)M2 * 4);
  int*   perm2    = (int*)alloc((size_t)M3 * 4);
  int*   mapping1 = (int*)alloc((size_t)M1 * 4);
  int*   s1       = (int*)alloc((size_t)E_ * 4);
  int*   d1       = (int*)alloc((size_t)E_ * 4);
  int*   v1       = (int*)alloc((size_t)E_ * 4);
  float* pn       = (float*)alloc(2 * 4);
  float* xs       = (float*)alloc((size_t)T_ * 128 * 128 * 4);
  float* gi_f     = (float*)alloc((size_t)T_ * 128 * 360 * 4);
  float* gi_b     = (float*)alloc((size_t)T_ * 128 * 360 * 4);
  float* hA       = (float*)alloc((size_t)128 * H_ * 4);
  float* hB       = (float*)alloc((size_t)128 * H_ * 4);
  float* hC       = (float*)alloc((size_t)128 * H_ * 4);
  float* hD       = (float*)alloc((size_t)128 * H_ * 4);
  float* xc       = (float*)alloc((size_t)128 * 4816 * 4);
  float* hid      = (float*)alloc((size_t)128 * 102 * 4);

  // ---- layer 1 ----
  hipMemsetAsync(agg1, 0, (size_t)M1 * 4 * 4, stream);
  hipMemsetAsync(mapping1, 0xFF, (size_t)M1 * 4, stream);
  k_scatter4<<<4096, 256, 0, stream>>>(x, esrc, edst, agg1);
  k_gconv1<<<(M1 / 16) / 8, 256, 0, stream>>>(agg1, x, W1_rel, W1_root, b1, h1, M1);
  k_pnorm<<<1, 128, 0, stream>>>(p1, pn);
  k_pnorm<<<1, 128, 0, stream>>>(p2, pn + 1);
  k_score<<<(M1 + 255) / 256, 256, 0, stream>>>(h1, p1, pn, score1, M1);
  k_topk<<<B_, 256, 0, stream>>>(score1, perm1, mapping1, N_, K1_);
  k_gather_scale<<<(M2 * 128 + 255) / 256, 256, 0, stream>>>(h1, score1, perm1, h1p, M2);
  k_readout<<<B_, 128, 0, stream>>>(h1p, xc, K1_, 0);
  k_remap<<<(E_ + 255) / 256, 256, 0, stream>>>(esrc, edst, mapping1, s1, d1, v1);

  // ---- layer 2 ----
  hipMemsetAsync(agg2, 0, (size_t)M2 * 128 * 4, stream);
  k_scatter128<<<8192, 256, 0, stream>>>(h1p, s1, d1, v1, agg2);
  k_gconv2<<<(M2 / 16) / 8, 256, 0, stream>>>(agg2, h1p, W2_rel, W2_root, b2, h2, M2);
  k_score<<<(M2 + 255) / 256, 256, 0, stream>>>(h2, p2, pn + 1, score2, M2);
  k_topk<<<B_, 256, 0, stream>>>(score2, perm2, (int*)nullptr, K1_, K2_);
  k_gather_scale<<<(M3 * 128 + 255) / 256, 256, 0, stream>>>(h2, score2, perm2, h2p, M3);
  k_readout<<<B_, 128, 0, stream>>>(h2p, xc, K2_, 1);

  // ---- target branch ----
  k_convpool<<<(T_ * 128 * 128 + 255) / 256, 256, 0, stream>>>(target, Wc, bc, xs);
  k_gi<<<(T_ * 128 * 360 + 255) / 256, 256, 0, stream>>>(xs, Wih_f, bih_f, gi_f);
  k_gi<<<(T_ * 128 * 360 + 255) / 256, 256, 0, stream>>>(xs, Wih_b, bih_b, gi_b);
  hipMemsetAsync(hA, 0, (size_t)128 * H_ * 4, stream);
  hipMemsetAsync(hC, 0, (size_t)128 * H_ * 4, stream);
  for (int t = 0; t < T_; ++t) {
    float* hin_f  = (t & 1) ? hB : hA;
    float* hout_f = (t & 1) ? hA : hB;
    float* hin_b  = (t & 1) ? hD : hC;
    float* hout_b = (t & 1) ? hC : hD;
    dim3 grid((128 * H_ + 255) / 256, 2);
    k_gru_step<<<grid, 256, 0, stream>>>(gi_f, gi_b, Whh_f, Whh_b, bhh_f, bhh_b,
                                         hin_f, hin_b, hout_f, hout_b, xc, t);
  }

  // ---- head ----
  k_dense1<<<(128 * 102 + 255) / 256, 256, 0, stream>>>(xc, Wd1, bd1, hid);
  k_out<<<1, 128, 0, stream>>>(hid, Wd3, bd3, out);
}